// GATStage4_AttentionSoftmax_Simple_51994874085813
// MI455X (gfx1250) — compile-verified
//
#include <hip/hip_runtime.h>
#include <stdint.h>

#define THREADS 256
#define EPT 8                      // edges per thread in pass 1
#define TILE (THREADS * EPT)       // 2048 edges per block

// ---- typed, address-space-qualified pointers for the async-LDS builtin ----
#if defined(__HIP_DEVICE_COMPILE__)
typedef int v4i __attribute__((vector_size(16)));            // v4i32, matches builtin proto
typedef __attribute__((address_space(1))) v4i gv4i;          // global
typedef __attribute__((address_space(3))) v4i lv4i;          // LDS

// Integer round-trip avoids C++ addrspace-cast restrictions.
static __device__ __forceinline__ gv4i* to_global_v4(const void* p) {
  return (gv4i*)(unsigned long long)(size_t)p;
}
// Generic shared-aperture address: low 32 bits ARE the LDS byte offset (ISA 10.2).
static __device__ __forceinline__ lv4i* to_lds_v4(void* p) {
  return (lv4i*)(unsigned int)(size_t)p;
}

#if __has_builtin(__builtin_amdgcn_global_load_async_to_lds_b128)
#define HAVE_ASYNC_LDS 1
#endif
#endif // __HIP_DEVICE_COMPILE__

// ---------------------------------------------------------------------------
// Kernel 0: zero the segment-sum accumulator (num_nodes lives on the device).
// ---------------------------------------------------------------------------
__global__ void __launch_bounds__(THREADS)
gat_zero_sums(float* __restrict__ sum_exp, const int* __restrict__ n_ptr) {
  const int n = *n_ptr;
  for (int i = blockIdx.x * blockDim.x + threadIdx.x; i < n;
       i += gridDim.x * blockDim.x) {
    sum_exp[i] = 0.0f;
  }
}

// ---------------------------------------------------------------------------
// Kernel 1: exp + segment-sum via global_atomic_add_f32 (L2-resident target).
// Edge tile staged through LDS with gfx1250 async loads when available.
// ---------------------------------------------------------------------------
__global__ void __launch_bounds__(THREADS)
gat_pass1_segsum(const float* __restrict__ e, const int* __restrict__ tgt,
                 float* __restrict__ sum_exp, int num_edges) {
  const int tid  = threadIdx.x;
  const int base = blockIdx.x * TILE;

#if defined(__HIP_DEVICE_COMPILE__) && defined(HAVE_ASYNC_LDS)
  __shared__ float s_e[TILE];
  __shared__ int   s_t[TILE];
  if (base + TILE <= num_edges) {
    // Each lane stages exactly the 8 edges it will process: 2x b128 per array.
    const float* ge = e   + base + tid * EPT;
    const int*   gt = tgt + base + tid * EPT;
    float* le = s_e + tid * EPT;
    int*   lt = s_t + tid * EPT;
    __builtin_amdgcn_global_load_async_to_lds_b128(to_global_v4(ge),     to_lds_v4(le),     0, 0);
    __builtin_amdgcn_global_load_async_to_lds_b128(to_global_v4(ge + 4), to_lds_v4(le + 4), 0, 0);
    __builtin_amdgcn_global_load_async_to_lds_b128(to_global_v4(gt),     to_lds_v4(lt),     0, 0);
    __builtin_amdgcn_global_load_async_to_lds_b128(to_global_v4(gt + 4), to_lds_v4(lt + 4), 0, 0);
#if __has_builtin(__builtin_amdgcn_s_wait_asynccnt)
    __builtin_amdgcn_s_wait_asynccnt(0);
#else
    asm volatile("s_wait_asynccnt 0" ::: "memory");
#endif
#pragma unroll
    for (int k = 0; k < EPT; ++k) {
      const float v = __expf(le[k]);
      atomicAdd(&sum_exp[lt[k]], v);   // global_atomic_add_f32, no return
    }
    return;
  }
#else
  if (base + TILE <= num_edges) {
    const float4* ge = (const float4*)(e   + base + tid * EPT);
    const int4*   gt = (const int4*)(tgt + base + tid * EPT);
    const float4 e0 = ge[0], e1 = ge[1];
    const int4   t0 = gt[0], t1 = gt[1];
    atomicAdd(&sum_exp[t0.x], __expf(e0.x));
    atomicAdd(&sum_exp[t0.y], __expf(e0.y));
    atomicAdd(&sum_exp[t0.z], __expf(e0.z));
    atomicAdd(&sum_exp[t0.w], __expf(e0.w));
    atomicAdd(&sum_exp[t1.x], __expf(e1.x));
    atomicAdd(&sum_exp[t1.y], __expf(e1.y));
    atomicAdd(&sum_exp[t1.z], __expf(e1.z));
    atomicAdd(&sum_exp[t1.w], __expf(e1.w));
    return;
  }
#endif
  // Tail (never taken for E = 6.4M, but keep it correct in general).
  for (int k = 0; k < EPT; ++k) {
    const int i = base + tid * EPT + k;
    if (i < num_edges) atomicAdd(&sum_exp[tgt[i]], __expf(e[i]));
  }
}

// ---------------------------------------------------------------------------
// Kernel 2: alpha = exp(e) / (sum_exp[tgt] + 1e-16). Re-reads of e/tgt hit L2
// (77 MB total working set << 192 MB L2); exp is recomputed (cheap VALU).
// ---------------------------------------------------------------------------
__global__ void __launch_bounds__(THREADS)
gat_pass2_normalize(const float* __restrict__ e, const int* __restrict__ tgt,
                    const float* __restrict__ sum_exp, float* __restrict__ out,
                    int num_edges) {
  int i = (blockIdx.x * THREADS + threadIdx.x) * 4;
  if (i + 4 <= num_edges) {
    const float4 ev = *(const float4*)(e + i);
    const int4   tv = *(const int4*)(tgt + i);
    float4 o;
    o.x = __expf(ev.x) / (sum_exp[tv.x] + 1e-16f);
    o.y = __expf(ev.y) / (sum_exp[tv.y] + 1e-16f);
    o.z = __expf(ev.z) / (sum_exp[tv.z] + 1e-16f);
    o.w = __expf(ev.w) / (sum_exp[tv.w] + 1e-16f);
    *(float4*)(out + i) = o;
  } else {
    for (; i < num_edges; ++i)
      out[i] = __expf(e[i]) / (sum_exp[tgt[i]] + 1e-16f);
  }
}

// ---------------------------------------------------------------------------
extern "C" void kernel_launch(void* const* d_in, const int* in_sizes, int n_in,
                              void* d_out, int out_size, void* d_ws, size_t ws_size,
                              hipStream_t stream) {
  const float* e          = (const float*)d_in[0];
  const int*   edge_index = (const int*)d_in[1];   // [2, E] row-major
  const int*   n_ptr      = (const int*)d_in[2];   // device scalar num_nodes

  const int E   = in_sizes[0];
  const int*   tgt = edge_index + E;               // edge_index[1]
  float* sum_exp = (float*)d_ws;                   // N floats of scratch
  float* alpha   = (float*)d_out;

  // 1) zero accumulator (grid-stride; N read on device)
  gat_zero_sums<<<512, THREADS, 0, stream>>>(sum_exp, n_ptr);

  // 2) exp + segment sum via float atomics
  const int g1 = (E + TILE - 1) / TILE;
  gat_pass1_segsum<<<g1, THREADS, 0, stream>>>(e, tgt, sum_exp, E);

  // 3) gather + normalize
  const int g2 = (E + THREADS * 4 - 1) / (THREADS * 4);
  gat_pass2_normalize<<<g2, THREADS, 0, stream>>>(e, tgt, sum_exp, alpha, E);
}